// Encoder_conv2_57655640982022
// MI455X (gfx1250) — compile-verified
//
#include <hip/hip_runtime.h>
#include <math.h>

// ---------------------------------------------------------------------------
// Problem constants (from the reference)
// ---------------------------------------------------------------------------
#define NLVL  5
#define NCELL 2
#define NCH   4

constexpr int kRes[NLVL]   = {8, 12, 20, 18, 32};
// ceil(r*r/16) 16-pixel tiles per level
constexpr int kTiles[NLVL] = {4, 9, 25, 21, 64};
// workspace layout: [level][cell][y][x][c] in fp32; offsets in float4 units
// per-level float4 count = 2*r*r : {128, 288, 800, 648, 2048}
constexpr int kBase4[NLVL] = {0, 128, 416, 1216, 1864};
constexpr int kTot4 = 3912;                       // 15648 floats = 62592 B
constexpr int kConvBlocks = (4 + 9 + 25 + 21 + 64) * (NCELL * NCH); // 984

typedef float v2f __attribute__((ext_vector_type(2)));
typedef float v8f __attribute__((ext_vector_type(8)));

// Async-to-LDS builtin parameter types (from clang diagnostic):
//   void (v4i addrspace(1)*, v4i addrspace(3)*, imm int, imm int)
typedef int v4i_gcc __attribute__((vector_size(16)));
typedef __attribute__((address_space(1))) v4i_gcc* glob_v4i_p;
typedef __attribute__((address_space(3))) v4i_gcc* lds_v4i_p;

// ---------------------------------------------------------------------------
// Kernel 1: a = tanh(depthwise_conv3x3(F)) via fp32 WMMA (16x16x4).
// One wave (32 threads) per (level, cell, channel, 16-pixel tile).
//   A (16x4 f32): 16 pixels x 4 im2col taps; 3 chunks cover taps 0..8 (+pad).
//   B (4x16 f32): filter taps broadcast across all 16 columns.
//   D column 0 = conv result for the 16 pixels.
// Output written channel-interleaved: ws[lvl][cell][pix][ch].
// ---------------------------------------------------------------------------
__global__ __launch_bounds__(32) void conv_tanh_wmma(
    const float* __restrict__ cw,
    const float* __restrict__ F0, const float* __restrict__ F1,
    const float* __restrict__ F2, const float* __restrict__ F3,
    const float* __restrict__ F4,
    float* __restrict__ ws)
{
  int rem = blockIdx.x;
  int lvl = 0;
  while (lvl < NLVL - 1 && rem >= kTiles[lvl] * (NCELL * NCH)) {
    rem -= kTiles[lvl] * (NCELL * NCH);
    ++lvl;
  }
  const int R     = kRes[lvl];
  const int npix  = R * R;
  const int tiles = kTiles[lvl];
  const int cc    = rem / tiles;   // 0..7 = cell*4 + ch
  const int tile  = rem % tiles;
  const int cell  = cc >> 2;
  const int ch    = cc & 3;

  const float* F  = (lvl == 0) ? F0 : (lvl == 1) ? F1 : (lvl == 2) ? F2
                  : (lvl == 3) ? F3 : F4;
  const float* Fc = F + (cell * NCH + ch) * npix;   // [R][R] plane

  float w9[9];
#pragma unroll
  for (int i = 0; i < 9; ++i) w9[i] = cw[ch * 9 + i];

  const int lane  = threadIdx.x;
  const int m     = lane & 15;     // A-matrix row (pixel in tile)
  const int khalf = lane >> 4;     // 0: K=0,1  1: K=2,3  (ISA A layout)
  const int p     = tile * 16 + m;
  const bool valid = p < npix;
  const int py = p / R;
  const int px = p % R;

  auto patch = [&](int tap) -> float {   // im2col with SAME zero padding
    if (tap >= 9 || !valid) return 0.0f;
    int dy = tap / 3 - 1;
    int dx = tap % 3 - 1;
    int yy = py + dy, xx = px + dx;
    if (yy < 0 || yy >= R || xx < 0 || xx >= R) return 0.0f;
    return Fc[yy * R + xx];
  };

#if __has_builtin(__builtin_amdgcn_wmma_f32_16x16x4_f32)
  v8f acc = {};
#pragma unroll
  for (int j = 0; j < 3; ++j) {          // K-chunks: taps [4j, 4j+3]
    const int tb = j * 4;
    v2f a, b;
    const int t = tb + khalf * 2;
    a.x = patch(t);
    a.y = patch(t + 1);
    // B mirrors the A K-layout: vgpr0 = K0(lo)/K2(hi), vgpr1 = K1(lo)/K3(hi),
    // same value for all 16 columns (broadcast).
    const float b0 = (tb + 0 < 9) ? w9[tb + 0] : 0.0f;
    const float b1 = (tb + 1 < 9) ? w9[tb + 1] : 0.0f;
    const float b2 = (tb + 2 < 9) ? w9[tb + 2] : 0.0f;
    const float b3 = (tb + 3 < 9) ? w9[tb + 3] : 0.0f;
    b.x = khalf ? b2 : b0;
    b.y = khalf ? b3 : b1;
    acc = __builtin_amdgcn_wmma_f32_16x16x4_f32(
        /*neg_a=*/false, a, /*neg_b=*/false, b,
        /*c_mod=*/(short)0, acc, /*reuse_a=*/false, /*reuse_b=*/false);
  }
  // D layout: VGPR v holds M=v (lanes 0-15) / M=v+8 (lanes 16-31), N=lane%16.
  // Column 0 lives in lanes 0 and 16.
  if (m == 0) {
#pragma unroll
    for (int v = 0; v < 8; ++v) {
      const int mm = v + khalf * 8;
      const int pp = tile * 16 + mm;
      if (pp < npix)
        ws[kBase4[lvl] * 4 + (cell * npix + pp) * NCH + ch] = tanhf(acc[v]);
    }
  }
#else
  // Scalar fallback (should not trigger on gfx1250 toolchains)
  if (khalf == 0 && valid) {
    float s = 0.0f;
#pragma unroll
    for (int t = 0; t < 9; ++t) s += patch(t) * w9[t];
    ws[kBase4[lvl] * 4 + (cell * npix + p) * NCH + ch] = tanhf(s);
  }
#endif
}

// ---------------------------------------------------------------------------
// Kernel 2: bilinear grid-sample of 1M points against LDS-resident features.
// ---------------------------------------------------------------------------
#define BLK 256
#define PPT 4

template <int R, int B4>
__device__ __forceinline__ float4 sample_level(const float4* __restrict__ sm,
                                               float gx, float gy)
{
  const float hs = 0.5f * (float)(R - 1);
  const float ix = (gx + 1.0f) * hs;
  const float iy = (gy + 1.0f) * hs;
  float a0 = 0.f, a1 = 0.f, a2 = 0.f, a3 = 0.f;
#pragma unroll
  for (int n = 0; n < NCELL; ++n) {
    const float fx  = ix + (float)n * 0.5f;      // off = n / NCELLS
    const float fy  = iy + (float)n * 0.5f;
    const float fx0 = floorf(fx);
    const float fy0 = floorf(fy);
    const float wx  = fx - fx0;
    const float wy  = fy - fy0;
    const int x0 = (int)fx0;
    const int y0 = (int)fy0;
    const int x0c = min(max(x0,     0), R - 1);
    const int x1c = min(max(x0 + 1, 0), R - 1);
    const int y0c = min(max(y0,     0), R - 1);
    const int y1c = min(max(y0 + 1, 0), R - 1);
    const float4* cb = sm + (B4 + n * R * R);
    const float4 nw = cb[y0c * R + x0c];   // ds_load_b128: 4 channels at once
    const float4 ne = cb[y0c * R + x1c];
    const float4 sw = cb[y1c * R + x0c];
    const float4 se = cb[y1c * R + x1c];
    const float wnw = (1.f - wx) * (1.f - wy);
    const float wne = wx * (1.f - wy);
    const float wsw = (1.f - wx) * wy;
    const float wse = wx * wy;
    a0 += nw.x * wnw + ne.x * wne + sw.x * wsw + se.x * wse;
    a1 += nw.y * wnw + ne.y * wne + sw.y * wsw + se.y * wse;
    a2 += nw.z * wnw + ne.z * wne + sw.z * wsw + se.z * wse;
    a3 += nw.w * wnw + ne.w * wne + sw.w * wsw + se.w * wse;
  }
  return make_float4(a0, a1, a2, a3);
}

__global__ __launch_bounds__(BLK) void sample_kernel(
    const float* __restrict__ xs, const float* __restrict__ ys,
    const float* __restrict__ ws, float* __restrict__ out, int npts)
{
  __shared__ float4 smem[kTot4];               // 62592 B of LDS
  const float4* g = (const float4*)ws;

#if __has_builtin(__builtin_amdgcn_global_load_async_to_lds_b128)
  for (int i = threadIdx.x; i < kTot4; i += BLK) {
    __builtin_amdgcn_global_load_async_to_lds_b128(
        (glob_v4i_p)(g + i),       // C-style cast: strips const, AS1, v4i
        (lds_v4i_p)(smem + i),     // generic->AS3 addrspacecast
        /*offset=*/0, /*cpol=*/0);
  }
  asm volatile("s_wait_asynccnt 0" ::: "memory");
#else
  for (int i = threadIdx.x; i < kTot4; i += BLK) smem[i] = g[i];
#endif
  __syncthreads();

  const int p0 = blockIdx.x * (BLK * PPT) + threadIdx.x;
#pragma unroll
  for (int it = 0; it < PPT; ++it) {
    const int p = p0 + it * BLK;
    if (p < npts) {
      const float gx = xs[p] * 2.0f - 1.0f;     // domain (0,1) -> (-1,1)
      const float gy = ys[p] * 2.0f - 1.0f;
      float4* orow = (float4*)(out + (size_t)p * 20);  // 80 B rows, 16 B aligned
      orow[0] = sample_level< 8,    0>(smem, gx, gy);
      orow[1] = sample_level<12,  128>(smem, gx, gy);
      orow[2] = sample_level<20,  416>(smem, gx, gy);
      orow[3] = sample_level<18, 1216>(smem, gx, gy);
      orow[4] = sample_level<32, 1864>(smem, gx, gy);
    }
  }
}

// ---------------------------------------------------------------------------
// Launch
// ---------------------------------------------------------------------------
extern "C" void kernel_launch(void* const* d_in, const int* in_sizes, int n_in,
                              void* d_out, int out_size, void* d_ws, size_t ws_size,
                              hipStream_t stream)
{
  // setup_inputs() order: x, y, conv_w, F0, F1, F2, F3, F4
  const float* x  = (const float*)d_in[0];
  const float* y  = (const float*)d_in[1];
  const float* cw = (const float*)d_in[2];
  const float* F0 = (const float*)d_in[3];
  const float* F1 = (const float*)d_in[4];
  const float* F2 = (const float*)d_in[5];
  const float* F3 = (const float*)d_in[6];
  const float* F4 = (const float*)d_in[7];
  float* ws  = (float*)d_ws;
  float* out = (float*)d_out;
  const int npts = in_sizes[0];

  conv_tanh_wmma<<<kConvBlocks, 32, 0, stream>>>(cw, F0, F1, F2, F3, F4, ws);

  const int blocks = (npts + BLK * PPT - 1) / (BLK * PPT);
  sample_kernel<<<blocks, BLK, 0, stream>>>(x, y, ws, out, npts);
}